// GraphConvBlock_52965536694818
// MI455X (gfx1250) — compile-verified
//
#include <hip/hip_runtime.h>
#include <hip/hip_bf16.h>
#include <stdint.h>

// Problem constants (match reference)
#define BATCH   2
#define CIN     128
#define OUTC    128
#define TLEN    25000
#define NNODES  (BATCH * TLEN)     // 50000 (divisible by 16)
#define NEDGES  800000

typedef float v2f __attribute__((ext_vector_type(2)));
typedef float v8f __attribute__((ext_vector_type(8)));

// ---------------------------------------------------------------------------
// 0) zero workspace region (agg + deg) -- grid-stride, graph-capture safe
// ---------------------------------------------------------------------------
__global__ void zero_kernel(float* p, long long n) {
    long long i = (long long)blockIdx.x * blockDim.x + threadIdx.x;
    long long stride = (long long)gridDim.x * blockDim.x;
    for (; i < n; i += stride) p[i] = 0.0f;
}

// ---------------------------------------------------------------------------
// 1) in-degree via scatter-add of 1.0 at dst
// ---------------------------------------------------------------------------
__global__ void degree_kernel(const long long* __restrict__ edge_index,
                              float* __restrict__ deg) {
    int e = blockIdx.x * blockDim.x + threadIdx.x;
    if (e >= NEDGES) return;
    long long dst = edge_index[NEDGES + e];
    atomicAdd(&deg[dst], 1.0f);
}

// ---------------------------------------------------------------------------
// 2) dinv_sqrt[n] = rsqrt(deg+1), inv_deg[n] = 1/(deg+1)
// ---------------------------------------------------------------------------
__global__ void norm_kernel(const float* __restrict__ deg,
                            float* __restrict__ dinv,
                            float* __restrict__ invdeg) {
    int n = blockIdx.x * blockDim.x + threadIdx.x;
    if (n >= NNODES) return;
    float d = deg[n] + 1.0f;
    dinv[n]   = rsqrtf(d);
    invdeg[n] = 1.0f / d;
}

// ---------------------------------------------------------------------------
// 3) h = flat @ W via V_WMMA_F32_16X16X4_F32
//    flat[n][k] = x[b][k][t], n = b*TLEN + t
//    Block: 256 threads = 8 waves; wave w computes 16x16 tile at cols w*16.
//    Grid: NNODES/16 = 3125 blocks. No divergence -> EXEC all ones for WMMA.
// ---------------------------------------------------------------------------
__global__ __launch_bounds__(256) void gemm_wmma_kernel(
        const float* __restrict__ x,
        const float* __restrict__ W,
        float* __restrict__ h) {
    const int lane = threadIdx.x & 31;
    const int wave = threadIdx.x >> 5;          // 0..7 -> N strip
    const int rowBase = blockIdx.x * 16;        // M tile base (node rows)
    const int m   = lane & 15;                  // row within tile (A) / col (B,D)
    const int hi  = lane >> 4;                  // half-wave select

    // A source: x[b][k][t] with row = rowBase + m
    const int row = rowBase + m;
    const int b   = row / TLEN;
    const int t   = row - b * TLEN;
    const float* xrow = x + (size_t)b * CIN * TLEN + t;   // index by k*TLEN

    // B source: W[k][n] with n = wave*16 + (lane&15)
    const int ncol = wave * 16 + m;

    v8f acc = {};
    #pragma unroll 4
    for (int k0 = 0; k0 < CIN; k0 += 4) {
        const int ka = k0 + 2 * hi;             // lanes 0-15: K=k0,k0+1 ; 16-31: K=k0+2,k0+3
        v2f afrag, bfrag;
        afrag.x = xrow[(size_t)(ka    ) * TLEN];
        afrag.y = xrow[(size_t)(ka + 1) * TLEN];
        bfrag.x = W[(ka    ) * OUTC + ncol];
        bfrag.y = W[(ka + 1) * OUTC + ncol];
        acc = __builtin_amdgcn_wmma_f32_16x16x4_f32(
                  /*neg_a=*/false, afrag,
                  /*neg_b=*/false, bfrag,
                  /*c_mod=*/(short)0, acc,
                  /*reuse_a=*/false, /*reuse_b=*/false);
    }

    // D layout: lane holds col ncol, rows (hi*8 .. hi*8+7)
    const int mbase = hi * 8;
    float* hp = h + (size_t)(rowBase + mbase) * OUTC + ncol;
    #pragma unroll
    for (int i = 0; i < 8; ++i)
        hp[(size_t)i * OUTC] = acc[i];
}

// ---------------------------------------------------------------------------
// 4) edge scatter: one wave32 per edge; lane handles 4 channels (float4 gather,
//    4x global_atomic_add_f32). h row = 512B -> wave gather fully coalesced.
// ---------------------------------------------------------------------------
__global__ __launch_bounds__(256) void edge_scatter_kernel(
        const float* __restrict__ h,
        const long long* __restrict__ edge_index,
        const float* __restrict__ dinv,
        float* __restrict__ agg) {
    const int gid  = blockIdx.x * blockDim.x + threadIdx.x;
    const int e    = gid >> 5;                  // wave id == edge id
    const int lane = threadIdx.x & 31;
    if (e >= NEDGES) return;

    const long long src = edge_index[e];
    const long long dst = edge_index[NEDGES + e];
    const float norm = dinv[src] * dinv[dst];

    const float4 v = ((const float4*)(h + (size_t)src * OUTC))[lane];
    float* ap = agg + (size_t)dst * OUTC + lane * 4;
    atomicAdd(ap + 0, v.x * norm);
    atomicAdd(ap + 1, v.y * norm);
    atomicAdd(ap + 2, v.z * norm);
    atomicAdd(ap + 3, v.w * norm);
}

// ---------------------------------------------------------------------------
// 5) finalize: out[b][oc][t] = relu(agg[n][oc] + h[n][oc]*invdeg[n] + bias[oc])
// ---------------------------------------------------------------------------
__global__ void finalize_kernel(const float* __restrict__ agg,
                                const float* __restrict__ h,
                                const float* __restrict__ invdeg,
                                const float* __restrict__ bias,
                                float* __restrict__ out) {
    long long idx = (long long)blockIdx.x * blockDim.x + threadIdx.x;
    const long long total = (long long)BATCH * OUTC * TLEN;
    if (idx >= total) return;
    const int t  = (int)(idx % TLEN);
    const int oc = (int)((idx / TLEN) % OUTC);
    const int b  = (int)(idx / ((long long)TLEN * OUTC));
    const long long n = (long long)b * TLEN + t;
    float val = agg[n * OUTC + oc] + h[n * OUTC + oc] * invdeg[n] + bias[oc];
    out[idx] = fmaxf(val, 0.0f);
}

// ---------------------------------------------------------------------------
// launcher
// ---------------------------------------------------------------------------
extern "C" void kernel_launch(void* const* d_in, const int* in_sizes, int n_in,
                              void* d_out, int out_size, void* d_ws, size_t ws_size,
                              hipStream_t stream) {
    const float*      x    = (const float*)d_in[0];      // [B, C, T]
    const float*      W    = (const float*)d_in[1];      // [C, OUT_C]
    const float*      bias = (const float*)d_in[2];      // [OUT_C]
    const long long*  ei   = (const long long*)d_in[3];  // [2, E] int64
    float* out = (float*)d_out;

    // workspace layout
    float* h      = (float*)d_ws;                        // N*128
    float* agg    = h      + (size_t)NNODES * OUTC;      // N*128
    float* deg    = agg    + (size_t)NNODES * OUTC;      // N
    float* dinv   = deg    + NNODES;                     // N
    float* invdeg = dinv   + NNODES;                     // N

    // 0) zero agg + deg (contiguous region: agg .. deg end)
    const long long nzero = (long long)NNODES * OUTC + NNODES;
    zero_kernel<<<2048, 256, 0, stream>>>(agg, nzero);

    // 1) degree
    degree_kernel<<<(NEDGES + 255) / 256, 256, 0, stream>>>(ei, deg);

    // 2) norms
    norm_kernel<<<(NNODES + 255) / 256, 256, 0, stream>>>(deg, dinv, invdeg);

    // 3) GEMM h = flat @ W (WMMA)
    gemm_wmma_kernel<<<NNODES / 16, 256, 0, stream>>>(x, W, h);

    // 4) edge gather/scale/scatter-add (1 wave per edge)
    {
        const long long threads = (long long)NEDGES * 32;
        const int blocks = (int)((threads + 255) / 256);
        edge_scatter_kernel<<<blocks, 256, 0, stream>>>(h, ei, dinv, agg);
    }

    // 5) fused self-loop + bias + relu + layout transpose
    {
        const long long total = (long long)BATCH * OUTC * TLEN;
        finalize_kernel<<<(int)((total + 255) / 256), 256, 0, stream>>>(agg, h, invdeg, bias, out);
    }
}